// Net_58385785422172
// MI455X (gfx1250) — compile-verified
//
#include <hip/hip_runtime.h>
#include <math.h>

// ---------------------------------------------------------------------------
// 3-layer GCN forward for MI455X (gfx1250, wave32).
// Dense projections use V_WMMA_F32_16X16X4_F32 (one wave per 16-row tile).
// Edge propagation uses fire-and-forget global_atomic_add_f32; the 6.4 MB
// feature array lives in the 192 MB L2, so gather+scatter stays on-chip.
// ---------------------------------------------------------------------------

typedef __attribute__((ext_vector_type(2))) float v2f;
typedef __attribute__((ext_vector_type(8))) float v8f;

static __device__ __forceinline__ void atomic_add_f32(float* p, float v) {
#if defined(__gfx1250__)
  unsafeAtomicAdd(p, v);           // global_atomic_add_f32 (no return)
#else
  atomicAdd(p, v);
#endif
}

// ------------------------------ utility kernels ----------------------------

__global__ void fill_f32(float* __restrict__ p, int n, float v) {
  int i = blockIdx.x * blockDim.x + threadIdx.x;
  if (i < n) p[i] = v;
}

// deg[col[e]] += ew[e]   (deg pre-initialized to 1.0 = self-loop weight)
__global__ void deg_scatter(const int* __restrict__ col, const float* __restrict__ ew,
                            float* __restrict__ deg, int E) {
  int e = blockIdx.x * blockDim.x + threadIdx.x;
  if (e < E) atomic_add_f32(&deg[col[e]], ew[e]);
}

// deg -> deg^{-1/2} in place
__global__ void deg_to_dinv(float* __restrict__ deg, int n) {
  int i = blockIdx.x * blockDim.x + threadIdx.x;
  if (i < n) {
    float d = deg[i];
    deg[i] = (d > 0.0f) ? rsqrtf(d) : 0.0f;
  }
}

// norm[e] = dinv[row[e]] * ew[e] * dinv[col[e]]
__global__ void edge_norm(const int* __restrict__ row, const int* __restrict__ col,
                          const float* __restrict__ ew, const float* __restrict__ dinv,
                          float* __restrict__ nrm, int E) {
  int e = blockIdx.x * blockDim.x + threadIdx.x;
  if (e < E) nrm[e] = dinv[row[e]] * ew[e] * dinv[col[e]];
}

// ------------------------------ WMMA GEMM ----------------------------------
// T[N,16] = X[N,K] @ W[K,16], fp32, via V_WMMA_F32_16X16X4_F32.
// One wave handles one 16-row tile; K/4 WMMA steps accumulate in 8 VGPRs.
// ISA layouts (wave32):
//   A 16x4 : lanes 0-15 -> M=lane, {K0,K0+1}; lanes 16-31 -> M=lane-16, {K0+2,K0+3}
//   B 4x16 : VGPR r, lanes 0-15 -> K=K0+r, N=lane; lanes 16-31 -> K=K0+2+r, N=lane-16
//   C 16x16: VGPR r, lanes 0-15 -> M=r, N=lane; lanes 16-31 -> M=r+8, N=lane-16
__global__ void gemm_wmma_n16(const float* __restrict__ X, const float* __restrict__ W,
                              float* __restrict__ T, int nTiles, int K) {
  int wave = (int)((blockIdx.x * blockDim.x + threadIdx.x) >> 5);
  int lane = threadIdx.x & 31;
  if (wave >= nTiles) return;                 // wave-uniform exit: EXEC stays all-1s
  const int half = lane >> 4;                 // 0: lanes 0-15, 1: lanes 16-31
  const int l    = lane & 15;
  const long long row0 = (long long)wave * 16;

  v8f acc = {};
  for (int k0 = 0; k0 < K; k0 += 4) {
    v2f a, b;
    const float* xr = X + (row0 + l) * K + (k0 + half * 2);
    a.x = xr[0];
    a.y = xr[1];
    const float* wr = W + (long long)(k0 + half * 2) * 16 + l;
    b.x = wr[0];
    b.y = wr[16];
    acc = __builtin_amdgcn_wmma_f32_16x16x4_f32(
        /*neg_a=*/false, a, /*neg_b=*/false, b,
        /*c_mod=*/(short)0, acc, /*reuse_a=*/false, /*reuse_b=*/false);
  }

  float* out = T + (row0 + half * 8) * 16 + l;
#pragma unroll
  for (int r = 0; r < 8; ++r) out[(long long)r * 16] = acc[r];
}

// scalar tail for N % 16 rows (not hit for N=100000, kept for safety)
__global__ void gemm_tail_n16(const float* __restrict__ X, const float* __restrict__ W,
                              float* __restrict__ T, int rowStart, int nRows, int K) {
  int idx = blockIdx.x * blockDim.x + threadIdx.x;
  if (idx >= nRows * 16) return;
  int r = rowStart + (idx >> 4), c = idx & 15;
  float s = 0.0f;
  for (int k = 0; k < K; ++k) s += X[(long long)r * K + k] * W[(long long)k * 16 + c];
  T[(long long)r * 16 + c] = s;
}

// T3[N,4] = H[N,16] @ W2[16,4]  (tiny; VALU is fine)
__global__ void gemm16x4(const float* __restrict__ H, const float* __restrict__ W2,
                         float* __restrict__ T3, int n) {
  int idx = blockIdx.x * blockDim.x + threadIdx.x;
  if (idx >= n * 4) return;
  int i = idx >> 2, c = idx & 3;
  const float* h = H + (long long)i * 16;
  float s = 0.0f;
#pragma unroll
  for (int k = 0; k < 16; ++k) s += h[k] * W2[k * 4 + c];
  T3[idx] = s;
}

// ------------------------------ propagation --------------------------------
// agg[col[e]*F + f] += norm[e] * t[row[e]*F + f]
// Thread = (edge, feature): 16 (or 4) consecutive lanes hit 16 consecutive
// atomic addresses -> coalesced L2 atomic bursts.
template <int F>
__global__ void scatter_feat(const int* __restrict__ row, const int* __restrict__ col,
                             const float* __restrict__ nrm, const float* __restrict__ t,
                             float* __restrict__ agg, int E) {
  int idx = blockIdx.x * blockDim.x + threadIdx.x;
  if (idx >= E * F) return;
  int e = idx / F, f = idx % F;    // F is a power of two -> shifts
  float nv = nrm[e];
  int r = row[e], c = col[e];
  atomic_add_f32(&agg[(long long)c * F + f], nv * t[(long long)r * F + f]);
}

// h[i,f] = act(agg[i,f] + dinv[i]^2 * t[i,f] + b[f]), written in place into t
template <int F, bool RELU>
__global__ void combine_self(const float* __restrict__ agg, float* __restrict__ t_inout,
                             const float* __restrict__ dinv, const float* __restrict__ bias,
                             int n) {
  int idx = blockIdx.x * blockDim.x + threadIdx.x;
  if (idx >= n * F) return;
  int i = idx / F, f = idx % F;
  float di = dinv[i];
  float v = agg[idx] + di * di * t_inout[idx] + bias[f];
  if (RELU) v = fmaxf(v, 0.0f);
  t_inout[idx] = v;
}

// out[i,:] = log_softmax(agg3[i,:] + dinv[i]^2 * t3[i,:] + b2)
__global__ void finalize_logsoftmax(const float* __restrict__ agg3, const float* __restrict__ t3,
                                    const float* __restrict__ dinv, const float* __restrict__ b2,
                                    float* __restrict__ out, int n) {
  int i = blockIdx.x * blockDim.x + threadIdx.x;
  if (i >= n) return;
  float d2 = dinv[i] * dinv[i];
  float v[4];
  float m = -__builtin_inff();
#pragma unroll
  for (int c = 0; c < 4; ++c) {
    v[c] = agg3[(long long)i * 4 + c] + d2 * t3[(long long)i * 4 + c] + b2[c];
    m = fmaxf(m, v[c]);
  }
  float s = 0.0f;
#pragma unroll
  for (int c = 0; c < 4; ++c) s += __expf(v[c] - m);
  float lse = m + __logf(s);
#pragma unroll
  for (int c = 0; c < 4; ++c) out[(long long)i * 4 + c] = v[c] - lse;
}

// ------------------------------ launcher -----------------------------------

extern "C" void kernel_launch(void* const* d_in, const int* in_sizes, int n_in,
                              void* d_out, int out_size, void* d_ws, size_t ws_size,
                              hipStream_t stream) {
  const float* x  = (const float*)d_in[0];
  const int*   ei = (const int*)d_in[1];   // [2, E] int32 (JAX x64 off)
  const float* ew = (const float*)d_in[2];
  const float* W1 = (const float*)d_in[3];
  const float* b1 = (const float*)d_in[4];
  const float* W3 = (const float*)d_in[5];
  const float* b3 = (const float*)d_in[6];
  const float* W2 = (const float*)d_in[7];
  const float* b2 = (const float*)d_in[8];
  float* out = (float*)d_out;

  const int FIN = 64;
  const int N = in_sizes[0] / FIN;
  const int E = in_sizes[2];
  const int* row = ei;        // sources
  const int* col = ei + E;    // targets

  // workspace layout (floats): dinv[N] | norm[E] | bufA[N*16] | bufB[N*16] | t3[N*4] | agg3[N*4]
  float* ws   = (float*)d_ws;
  float* dinv = ws;
  float* nrm  = dinv + N;
  float* bufA = nrm + E;
  float* bufB = bufA + (size_t)N * 16;
  float* t3   = bufB + (size_t)N * 16;
  float* agg3 = t3 + (size_t)N * 4;

  const int TB = 256;
  auto nb = [](long long n, int tb) { return dim3((unsigned)((n + tb - 1) / tb)); };

  // --- degree / normalization ---
  fill_f32<<<nb(N, TB), TB, 0, stream>>>(dinv, N, 1.0f);           // self-loop weight
  deg_scatter<<<nb(E, TB), TB, 0, stream>>>(col, ew, dinv, E);
  deg_to_dinv<<<nb(N, TB), TB, 0, stream>>>(dinv, N);
  edge_norm<<<nb(E, TB), TB, 0, stream>>>(row, col, ew, dinv, nrm, E);

  const int nTiles  = N / 16;
  const int remRows = N - nTiles * 16;
  const long long waveThreads = (long long)nTiles * 32;

  // --- layer 1: t = x @ W1 -> bufA ; agg -> bufB ; h1 -> bufA ---
  gemm_wmma_n16<<<nb(waveThreads, TB), TB, 0, stream>>>(x, W1, bufA, nTiles, FIN);
  if (remRows)
    gemm_tail_n16<<<nb((long long)remRows * 16, TB), TB, 0, stream>>>(x, W1, bufA, nTiles * 16, remRows, FIN);
  fill_f32<<<nb((long long)N * 16, TB), TB, 0, stream>>>(bufB, N * 16, 0.0f);
  scatter_feat<16><<<nb((long long)E * 16, TB), TB, 0, stream>>>(row, col, nrm, bufA, bufB, E);
  combine_self<16, true><<<nb((long long)N * 16, TB), TB, 0, stream>>>(bufB, bufA, dinv, b1, N);

  // --- layer 2: t2 = h1 @ W3 -> bufB ; agg -> bufA ; h2 -> bufB ---
  gemm_wmma_n16<<<nb(waveThreads, TB), TB, 0, stream>>>(bufA, W3, bufB, nTiles, 16);
  if (remRows)
    gemm_tail_n16<<<nb((long long)remRows * 16, TB), TB, 0, stream>>>(bufA, W3, bufB, nTiles * 16, remRows, 16);
  fill_f32<<<nb((long long)N * 16, TB), TB, 0, stream>>>(bufA, N * 16, 0.0f);
  scatter_feat<16><<<nb((long long)E * 16, TB), TB, 0, stream>>>(row, col, nrm, bufB, bufA, E);
  combine_self<16, true><<<nb((long long)N * 16, TB), TB, 0, stream>>>(bufA, bufB, dinv, b3, N);

  // --- layer 3: t3 = h2 @ W2 ; agg3 ; log_softmax -> out ---
  gemm16x4<<<nb((long long)N * 4, TB), TB, 0, stream>>>(bufB, W2, t3, N);
  fill_f32<<<nb((long long)N * 4, TB), TB, 0, stream>>>(agg3, N * 4, 0.0f);
  scatter_feat<4><<<nb((long long)E * 4, TB), TB, 0, stream>>>(row, col, nrm, t3, agg3, E);
  finalize_logsoftmax<<<nb(N, TB), TB, 0, stream>>>(agg3, t3, dinv, b2, out, N);
}